// ScreenShooting_42743514530384
// MI455X (gfx1250) — compile-verified
//
#include <hip/hip_runtime.h>
#include <cstdint>

#define BN 32
#define CC 3
#define HH 512
#define WW 512

typedef float v4f __attribute__((ext_vector_type(4)));

#if defined(__AMDGCN__)
#  if __has_builtin(__builtin_amdgcn_global_load_async_to_lds_b32)
#    define USE_ASYNC_LDS 1
#  endif
#  if __has_builtin(__builtin_amdgcn_s_wait_asynccnt)
#    define USE_WAIT_BUILTIN 1
#  endif
#endif

// ---------------------------------------------------------------------------
// Per-batch homography: DLT 8x8 solve (partial pivoting) -> adjugate of M.
// 32 threads total; cost is negligible. Ratios x/z are scale invariant so we
// store adj(M) = det*M^{-1} instead of dividing by det.
// ---------------------------------------------------------------------------
__global__ __launch_bounds__(32) void homog_kernel(const float* __restrict__ dsto,
                                                   float* __restrict__ minv) {
  int b = threadIdx.x;
  if (b >= BN) return;
  const float sx[4] = {0.f, 511.f, 511.f, 0.f};
  const float sy[4] = {0.f, 0.f, 511.f, 511.f};
  const float bx[4] = {0.f, 512.f, 512.f, 0.f};
  const float by[4] = {0.f, 0.f, 512.f, 512.f};
  float A[8][9];
  for (int k = 0; k < 4; ++k) {
    float u = bx[k] + dsto[(b * 4 + k) * 2 + 0];
    float v = by[k] + dsto[(b * 4 + k) * 2 + 1];
    float x = sx[k], y = sy[k];
    float* r1 = A[k];
    float* r2 = A[k + 4];
    r1[0] = x;   r1[1] = y;   r1[2] = 1.f; r1[3] = 0.f; r1[4] = 0.f; r1[5] = 0.f;
    r1[6] = -x * u; r1[7] = -y * u; r1[8] = u;
    r2[0] = 0.f; r2[1] = 0.f; r2[2] = 0.f; r2[3] = x;   r2[4] = y;   r2[5] = 1.f;
    r2[6] = -x * v; r2[7] = -y * v; r2[8] = v;
  }
  for (int col = 0; col < 8; ++col) {
    int piv = col;
    float best = fabsf(A[col][col]);
    for (int r = col + 1; r < 8; ++r) {
      float t = fabsf(A[r][col]);
      if (t > best) { best = t; piv = r; }
    }
    if (piv != col) {
      for (int j = col; j < 9; ++j) { float t = A[col][j]; A[col][j] = A[piv][j]; A[piv][j] = t; }
    }
    float inv = 1.0f / A[col][col];
    for (int r = col + 1; r < 8; ++r) {
      float f = A[r][col] * inv;
      for (int j = col; j < 9; ++j) A[r][j] -= f * A[col][j];
    }
  }
  float p[8];
  for (int r = 7; r >= 0; --r) {
    float s = A[r][8];
    for (int j = r + 1; j < 8; ++j) s -= A[r][j] * p[j];
    p[r] = s / A[r][r];
  }
  float m00 = p[0], m01 = p[1], m02 = p[2];
  float m10 = p[3], m11 = p[4], m12 = p[5];
  float m20 = p[6], m21 = p[7], m22 = 1.f;
  float o[9];
  o[0] =  (m11 * m22 - m12 * m21);
  o[1] = -(m01 * m22 - m02 * m21);
  o[2] =  (m01 * m12 - m02 * m11);
  o[3] = -(m10 * m22 - m12 * m20);
  o[4] =  (m00 * m22 - m02 * m20);
  o[5] = -(m00 * m12 - m02 * m10);
  o[6] =  (m10 * m21 - m11 * m20);
  o[7] = -(m00 * m21 - m01 * m20);
  o[8] =  (m00 * m11 - m01 * m10);
  for (int j = 0; j < 9; ++j) minv[b * 9 + j] = o[j];
}

// ---------------------------------------------------------------------------
// Light mask & moire helpers
// ---------------------------------------------------------------------------
__device__ __forceinline__ float light_at(float fh, float fw, float a, float bb,
                                          int lc, int d, float lx, float ly,
                                          float inv_ml) {
  if (lc == 0) {
    // linear gradient f(t) = -((b-a)/511)*(t-512)+a, rotated by direction
    float t = (d == 1) ? fh : (d == 2) ? fw : (d == 3) ? (511.0f - fh) : (511.0f - fw);
    return a - ((bb - a) * (1.0f / 511.0f)) * (t - 512.0f);
  }
  float dx = fh - lx, dy = fw - ly;
  return sqrtf(dx * dx + dy * dy) * inv_ml * (a - bb) + bb;
}

__device__ __forceinline__ float moire_at(float fh, float fw, float cx, float cy,
                                          float cth, float sth) {
  const float TP = 6.28318530717958647692f;
  float xg = fh + 1.0f, yg = fw + 1.0f;
  float dx = xg - cx, dy = yg - cy;
  float dist = sqrtf(dx * dx + dy * dy);
  float z1 = 0.5f + 0.5f * cosf(TP * dist);
  float z2 = 0.5f + 0.5f * cosf(TP * (cth * xg + sth * yg));
  return fminf(z1, z2);  // == _moire()*2-1 after algebraic simplification
}

__device__ __forceinline__ float inv_maxlen(float lx, float ly) {
  float m0 = sqrtf(lx * lx + ly * ly);
  float m1 = sqrtf((lx - 255.f) * (lx - 255.f) + ly * ly);
  float m2 = sqrtf(lx * lx + (ly - 255.f) * (ly - 255.f));
  float m3 = sqrtf((lx - 512.f) * (lx - 512.f) + (ly - 512.f) * (ly - 512.f));
  return 1.0f / fmaxf(fmaxf(m0, m1), fmaxf(m2, m3));
}

// ---------------------------------------------------------------------------
// Precompute L[H,W] (1 MB) and Zmin[3,H,W] (3 MB) -> stays L2-resident and is
// reused 32x by the main pass, instead of recomputing 2 cos + 2 sqrt per
// output element across the whole B*C*H*W domain.
// ---------------------------------------------------------------------------
__global__ __launch_bounds__(256) void lz_kernel(
    float* __restrict__ L, float* __restrict__ Z,
    const float* __restrict__ ab, const float* __restrict__ cent,
    const int* __restrict__ cflag, const int* __restrict__ dir,
    const int* __restrict__ lxy, const int* __restrict__ theta) {
  int w = blockIdx.x * 256 + threadIdx.x;
  int h = blockIdx.y;
  float fh = (float)h, fw = (float)w;
  float a = ab[0], bb = ab[1];
  int lc = cflag[0], d = dir[0];
  float lx = (float)lxy[0], ly = (float)lxy[1];
  float iml = inv_maxlen(lx, ly);
  L[h * WW + w] = light_at(fh, fw, a, bb, lc, d, lx, ly, iml);
  const float RAD = 0.01745329251994329577f;
  for (int ch = 0; ch < 3; ++ch) {
    float th = (float)theta[ch] * RAD;
    Z[(ch * HH + h) * WW + w] =
        moire_at(fh, fw, cent[2 * ch + 0], cent[2 * ch + 1], cosf(th), sinf(th));
  }
}

__device__ __forceinline__ float tap(const float* __restrict__ base, int xi, int yi) {
  bool ok = ((unsigned)xi < (unsigned)WW) & ((unsigned)yi < (unsigned)HH);
  int xc = min(max(xi, 0), WW - 1);
  int yc = min(max(yi, 0), HH - 1);
  float t = base[yc * WW + xc];
  return ok ? t : 0.0f;
}

// ---------------------------------------------------------------------------
// Main fused pass. block = (128,2) = 8 wave32s. Each thread produces 4
// consecutive output pixels (b128 NT store). Noise is read as b128 NT load
// (touched once -> keep it out of L2). Image gather uses regular temporal
// loads (bilinear reuse). Minv is broadcast per block through LDS using the
// gfx1250 async-to-LDS path.
// ---------------------------------------------------------------------------
template <bool PRECOMP>
__global__ __launch_bounds__(256) void ss_main(
    const float* __restrict__ img, const float* __restrict__ noise,
    const float* __restrict__ minv, const float* __restrict__ L,
    const float* __restrict__ Z, float* __restrict__ out,
    const float* __restrict__ ab, const float* __restrict__ cent,
    const int* __restrict__ cflag, const int* __restrict__ dir,
    const int* __restrict__ lxy, const int* __restrict__ theta) {
  const int bz = blockIdx.z;          // b*C + c
  const int b  = bz / CC;
  const int ch = bz - b * CC;
  const int h  = blockIdx.y * 2 + threadIdx.y;
  const int w0 = threadIdx.x * 4;

  __shared__ float sM[12];
  int lin = threadIdx.y * 128 + threadIdx.x;
#ifdef USE_ASYNC_LDS
  if (lin < 9) {
    const float* gp = minv + b * 9 + lin;
    __builtin_amdgcn_global_load_async_to_lds_b32(
        (__attribute__((address_space(1))) int*)(uintptr_t)gp,
        (__attribute__((address_space(3))) int*)(unsigned)(uintptr_t)&sM[lin],
        0, 0);
  }
#  ifdef USE_WAIT_BUILTIN
  __builtin_amdgcn_s_wait_asynccnt(0);
#  else
  asm volatile("s_wait_asynccnt 0" ::: "memory");
#  endif
  __syncthreads();
#else
  if (lin < 9) sM[lin] = minv[b * 9 + lin];
  __syncthreads();
#endif

  const float* base = img + (size_t)bz * (HH * WW);
  const size_t row = ((size_t)bz * HH + h) * WW + w0;
  const float fy = (float)h;

  // uniform scalar params (only needed for the fused fallback)
  float a = 0.f, bb = 0.f, lx = 0.f, ly = 0.f, iml = 0.f, cx = 0.f, cy = 0.f,
        cth = 0.f, sth = 0.f;
  int lc = 0, d = 1;
  if (!PRECOMP) {
    a = ab[0]; bb = ab[1];
    lc = cflag[0]; d = dir[0];
    lx = (float)lxy[0]; ly = (float)lxy[1];
    iml = inv_maxlen(lx, ly);
    cx = cent[2 * ch + 0]; cy = cent[2 * ch + 1];
    float th = (float)theta[ch] * 0.01745329251994329577f;
    cth = cosf(th); sth = sinf(th);
  }

  v4f nz = __builtin_nontemporal_load((const v4f*)(noise + row));
  v4f l4, z4;
  if (PRECOMP) {
    l4 = *(const v4f*)(L + h * WW + w0);
    z4 = *(const v4f*)(Z + (ch * HH + h) * WW + w0);
  }

  v4f res;
#pragma unroll
  for (int i = 0; i < 4; ++i) {
    const int x = w0 + i;
    const float fx = (float)x;
    float den = sM[6] * fx + sM[7] * fy + sM[8];
    float rd  = 1.0f / den;
    float xs = (sM[0] * fx + sM[1] * fy + sM[2]) * rd;
    float ys = (sM[3] * fx + sM[4] * fy + sM[5]) * rd;
    float x0f = floorf(xs), y0f = floorf(ys);
    int x0 = (int)x0f, y0 = (int)y0f;
    float wx = xs - x0f, wy = ys - y0f;
    float v00 = tap(base, x0,     y0);
    float v01 = tap(base, x0 + 1, y0);
    float v10 = tap(base, x0,     y0 + 1);
    float v11 = tap(base, x0 + 1, y0 + 1);
    float warped = v00 * (1.f - wx) * (1.f - wy) + v01 * wx * (1.f - wy) +
                   v10 * (1.f - wx) * wy + v11 * wx * wy;
    float Lv, Zv;
    if (PRECOMP) {
      Lv = l4[i];
      Zv = z4[i];
    } else {
      Lv = light_at(fy, fx, a, bb, lc, d, lx, ly, iml);
      Zv = moire_at(fy, fx, cx, cy, cth, sth);
    }
    res[i] = warped * Lv * 0.85f + Zv * 0.15f + 0.03162277660168379f * nz[i];
  }
  __builtin_nontemporal_store(res, (v4f*)(out + row));
}

// ---------------------------------------------------------------------------
extern "C" void kernel_launch(void* const* d_in, const int* in_sizes, int n_in,
                              void* d_out, int out_size, void* d_ws, size_t ws_size,
                              hipStream_t stream) {
  (void)in_sizes; (void)n_in; (void)out_size;
  const float* image = (const float*)d_in[0];   // [32,3,512,512] f32
  const float* dsto  = (const float*)d_in[1];   // [32,4,2] f32
  const float* ab    = (const float*)d_in[2];   // [2] f32
  const float* cent  = (const float*)d_in[3];   // [3,2] f32
  const float* noise = (const float*)d_in[4];   // [32,3,512,512] f32
  const int*   cflag = (const int*)d_in[5];     // [1] i32
  const int*   dir   = (const int*)d_in[6];     // [1] i32
  const int*   lxy   = (const int*)d_in[7];     // [2] i32
  const int*   theta = (const int*)d_in[8];     // [3] i32
  float* out = (float*)d_out;

  float* minv = (float*)d_ws;                           // 32*9 floats
  const size_t lz_off = 4096;
  float* L = (float*)((char*)d_ws + lz_off);            // H*W floats
  float* Z = L + (size_t)HH * WW;                       // 3*H*W floats
  const bool precomp = ws_size >= lz_off + 4ull * HH * WW * sizeof(float);

  homog_kernel<<<dim3(1), dim3(32), 0, stream>>>(dsto, minv);

  dim3 block(128, 2, 1);
  dim3 grid(1, HH / 2, BN * CC);
  if (precomp) {
    lz_kernel<<<dim3(WW / 256, HH), dim3(256), 0, stream>>>(L, Z, ab, cent, cflag,
                                                            dir, lxy, theta);
    ss_main<true><<<grid, block, 0, stream>>>(image, noise, minv, L, Z, out, ab,
                                              cent, cflag, dir, lxy, theta);
  } else {
    ss_main<false><<<grid, block, 0, stream>>>(image, noise, minv, L, Z, out, ab,
                                               cent, cflag, dir, lxy, theta);
  }
}